// Affine3D_18777597018143
// MI455X (gfx1250) — compile-verified
//
#include <hip/hip_runtime.h>

// ---------------------------------------------------------------------------
// 3D affine grid-sample (trilinear), B=2 C=4 H=W=D=128, float32.
//
// Coordinate transform Tg = theta(3x4) x grid(4xN) is done on the matrix unit
// with V_WMMA_F32_16X16X4_F32 (the reference's einsum). Interpolation is
// per-lane VALU with cached gathers; output uses non-temporal stores since it
// is write-once (keep L2 for the gather working set).
// ---------------------------------------------------------------------------

typedef __attribute__((ext_vector_type(2))) float v2f;
typedef __attribute__((ext_vector_type(8))) float v8f;

#define Hn 128
#define Wn 128
#define Dn 128
#define Cn 4
#define Bn 2
#define NPTS (Hn * Wn * Dn)   // 2,097,152 spatial points

__global__ __launch_bounds__(256) void affine3d_wmma_kernel(
    const float* __restrict__ x,      // (B,C,H,W,D)
    const float* __restrict__ theta,  // (1,3,4)
    float* __restrict__ out)          // (B,C,H,W,D)
{
    const int tid  = blockIdx.x * 256 + threadIdx.x;   // spatial point index p
    const int lane = threadIdx.x & 31;
    const int half = lane >> 4;        // 0: B rows K0/K1, 1: B rows K2/K3
    const int col  = lane & 15;        // WMMA column this lane feeds
    const int base = tid & ~31;        // first point of this wave

    // ---- A operand: theta zero-padded to 16x4 ------------------------------
    // Lane m (0..15) holds A[m][0..1]; lane 16+m holds A[m][2..3].
    v2f a; a[0] = 0.0f; a[1] = 0.0f;
    if (lane < 3)                 { a[0] = theta[lane * 4 + 0]; a[1] = theta[lane * 4 + 1]; }
    if (lane >= 16 && lane < 19)  { int m = lane - 16;
                                    a[0] = theta[m * 4 + 2];    a[1] = theta[m * 4 + 3]; }

    // ---- B operands: grid columns [xt, yt, zt, 1] for two 16-point tiles ---
    const float step = 2.0f / 127.0f;  // jnp.linspace(-1, 1, 128) step
    float xt0, yt0, zt0, xt1, yt1, zt1;
    {
        int p0 = base + col;
        int p1 = base + 16 + col;
        int i0 = p0 >> 14, j0 = (p0 >> 7) & 127, k0 = p0 & 127;
        int i1 = p1 >> 14, j1 = (p1 >> 7) & 127, k1 = p1 & 127;
        xt0 = fmaf((float)i0, step, -1.0f);
        yt0 = fmaf((float)j0, step, -1.0f);
        zt0 = fmaf((float)k0, step, -1.0f);
        xt1 = fmaf((float)i1, step, -1.0f);
        yt1 = fmaf((float)j1, step, -1.0f);
        zt1 = fmaf((float)k1, step, -1.0f);
    }
    // B(4x16) layout: VGPR0 = row K0 (lanes 0-15) / row K2 (lanes 16-31),
    //                 VGPR1 = row K1 / row K3.
    v2f b0, b1;
    b0[0] = half ? zt0 : xt0;  b0[1] = half ? 1.0f : yt0;
    b1[0] = half ? zt1 : xt1;  b1[1] = half ? 1.0f : yt1;

    v8f cz = {0.0f, 0.0f, 0.0f, 0.0f, 0.0f, 0.0f, 0.0f, 0.0f};
    // D rows 0..2 = xs, ys, zs for the 16 columns (lane n holds column n).
    v8f d0 = __builtin_amdgcn_wmma_f32_16x16x4_f32(false, a, false, b0,
                                                   (short)0, cz, false, false);
    v8f d1 = __builtin_amdgcn_wmma_f32_16x16x4_f32(false, a, false, b1,
                                                   (short)0, cz, false, false);

    // ---- redistribute: lane l (>=16) takes WMMA#1 results from lane l-16 ---
    float sx = __shfl_xor(d1[0], 16, 32);
    float sy = __shfl_xor(d1[1], 16, 32);
    float sz = __shfl_xor(d1[2], 16, 32);
    float xs = half ? sx : d0[0];
    float ys = half ? sy : d0[1];
    float zs = half ? sz : d0[2];

    // ---- trilinear setup (faithful to reference, incl. (z0f - zv) sign) ----
    float xc = fminf(fmaxf(xs, -1.0f), 1.0f);
    float yc = fminf(fmaxf(ys, -1.0f), 1.0f);
    float zc = fminf(fmaxf(zs, -1.0f), 1.0f);
    float xv = (xc + 1.0f) * 63.5f;   // indexes W
    float yv = (yc + 1.0f) * 63.5f;   // indexes H
    float zv = (zc + 1.0f) * 63.5f;   // indexes D

    float x0f = floorf(xv), y0f = floorf(yv), z0f = floorf(zv);
    int   x0 = (int)x0f,    y0 = (int)y0f,    z0 = (int)z0f;
    int   x1 = min(x0 + 1, 127);
    int   y1 = min(y0 + 1, 127);
    int   z1 = min(z0 + 1, 127);

    float fx = xv - x0f, fy = yv - y0f, fz = zv - z0f;
    float gx = 1.0f - fx, gy = 1.0f - fy;
    float gz0 = 1.0f - fz;   // (z1f - zv), weights wa..wd
    float gz1 = -fz;         // (z0f - zv), weights we..wh — reference's exact expr

    float c00 = gx * gy;     // (y0, x0)
    float c10 = gx * fy;     // (y1, x0)
    float c01 = fx * gy;     // (y0, x1)
    float c11 = fx * fy;     // (y1, x1)

    const int oy0x0 = y0 * (Wn * Dn) + x0 * Dn;
    const int oy1x0 = y1 * (Wn * Dn) + x0 * Dn;
    const int oy0x1 = y0 * (Wn * Dn) + x1 * Dn;
    const int oy1x1 = y1 * (Wn * Dn) + x1 * Dn;

    // ---- gather + blend for all (b, c); weights shared across 8 outputs ----
    #pragma unroll
    for (int b = 0; b < Bn; ++b) {
        #pragma unroll
        for (int c = 0; c < Cn; ++c) {
            const size_t chan = (size_t)(b * Cn + c) << 21;   // * H*W*D
            const float* src = x + chan;
            float a0 = src[oy0x0 + z0], a1 = src[oy0x0 + z1];
            float b0v = src[oy1x0 + z0], b1v = src[oy1x0 + z1];
            float c0 = src[oy0x1 + z0], c1 = src[oy0x1 + z1];
            float e0 = src[oy1x1 + z0], e1 = src[oy1x1 + z1];
            float r = c00 * fmaf(gz0, a0,  gz1 * a1)
                    + c10 * fmaf(gz0, b0v, gz1 * b1v)
                    + c01 * fmaf(gz0, c0,  gz1 * c1)
                    + c11 * fmaf(gz0, e0,  gz1 * e1);
            __builtin_nontemporal_store(r, out + chan + (size_t)tid);
        }
    }
}

extern "C" void kernel_launch(void* const* d_in, const int* in_sizes, int n_in,
                              void* d_out, int out_size, void* d_ws, size_t ws_size,
                              hipStream_t stream) {
    const float* x     = (const float*)d_in[0];
    const float* theta = (const float*)d_in[1];
    float*       out   = (float*)d_out;
    // Exactly NPTS threads (no tail guard -> EXEC all-ones for WMMA).
    dim3 grid(NPTS / 256), block(256);
    hipLaunchKernelGGL(affine3d_wmma_kernel, grid, block, 0, stream, x, theta, out);
}